// GroupingLayer_57157424775279
// MI455X (gfx1250) — compile-verified
//
#include <hip/hip_runtime.h>
#include <hip/hip_bf16.h>

// Problem dimensions from the reference (setup_inputs).
#define BATCH      8
#define N_POINTS   16384
#define NP         2048
#define N_FEAT     256
#define K_CLUSTER  64
#define N_CH       (N_FEAT + 3)     // 259
#define RAD2       0.16f            // 0.4^2
#define INV_RADIUS 2.5f             // 1/0.4
#define CHUNKS     8
#define PTS_PER_CHUNK (N_POINTS / CHUNKS)   // 2048

typedef __attribute__((ext_vector_type(2))) float v2f;
typedef __attribute__((ext_vector_type(8))) float v8f;

// ---------------------------------------------------------------------------
// Kernel 1: ball query via V_WMMA_F32_16X16X4_F32.
// One wave (32 threads) handles a 16-centroid tile against a 2048-point chunk.
//   D[m][n] = 2*c_m . p_n - |p_n|^2   (A row = (2cx,2cy,2cz,-1), B col = (px,py,pz,|p|^2))
//   dist^2 < r^2  <=>  D[m][n] > |c_m|^2 - r^2
// Running per-centroid max index, reduced across lanes, merged with atomicMax.
// ---------------------------------------------------------------------------
__global__ __launch_bounds__(32)
void ball_query_wmma(const float* __restrict__ pts,
                     const float* __restrict__ cents,
                     int* __restrict__ best_idx) {
    const int b     = blockIdx.y;
    const int tile  = blockIdx.x / CHUNKS;   // centroid tile (16 centroids)
    const int chunk = blockIdx.x % CHUNKS;
    const int lane  = threadIdx.x;           // 0..31, wave32
    const int half  = lane >> 4;             // 0: K=0,1   1: K=2,3
    const int lm    = lane & 15;

    __shared__ float thr[16];

    // Load this lane's centroid (m = lm within the tile).
    const float* cb = cents + ((size_t)b * NP + (size_t)tile * 16 + lm) * 3;
    const float cx = cb[0], cy = cb[1], cz = cb[2];
    if (half == 0) {
        thr[lm] = cx * cx + cy * cy + cz * cz - RAD2;   // |c|^2 - r^2
    }
    __syncthreads();

    // A-matrix fragment (16x4 f32): lane<16 -> (2cx,2cy); lane>=16 -> (2cz,-1)
    v2f a;
    a.x = half ? (2.0f * cz) : (2.0f * cx);
    a.y = half ? (-1.0f)     : (2.0f * cy);

    // Per-lane copy of the 8 thresholds this lane's D rows need (M = v + 8*half).
    float t[8];
#pragma unroll
    for (int v = 0; v < 8; ++v) t[v] = thr[v + 8 * half];

    int best[8];
#pragma unroll
    for (int v = 0; v < 8; ++v) best[v] = -1;

    const int pbase = chunk * PTS_PER_CHUNK;
    const float* pp = pts + (size_t)b * N_POINTS * 3;

    for (int t0 = 0; t0 < PTS_PER_CHUNK; t0 += 16) {
        const int n = pbase + t0 + lm;             // this lane's point column
        const float* pr = pp + (size_t)n * 3;
        const float px = pr[0], py = pr[1], pz = pr[2];
        const float pn2 = px * px + py * py + pz * pz;

        // B-matrix fragment (4x16 f32): lane<16 -> (px,py); lane>=16 -> (pz,|p|^2)
        v2f bm;
        bm.x = half ? pz  : px;
        bm.y = half ? pn2 : py;

        v8f cacc = {};
        // D = A x B + 0 : 16x16 tile of (2 c.p - |p|^2), one wmma per tile.
        v8f d = __builtin_amdgcn_wmma_f32_16x16x4_f32(
            /*neg_a=*/false, a, /*neg_b=*/false, bm,
            /*c_mod=*/(short)0, cacc, /*reuse_a=*/false, /*reuse_b=*/false);

#pragma unroll
        for (int v = 0; v < 8; ++v) {
            const int cand = (d[v] > t[v]) ? n : -1;   // strict < r^2
            best[v] = best[v] > cand ? best[v] : cand;
        }
    }

    // Reduce max across the 16 lanes sharing each centroid row (xor masks <= 8
    // stay inside each 16-lane half on wave32).
#pragma unroll
    for (int v = 0; v < 8; ++v) {
        int x = best[v];
        x = max(x, __shfl_xor(x, 1, 32));
        x = max(x, __shfl_xor(x, 2, 32));
        x = max(x, __shfl_xor(x, 4, 32));
        x = max(x, __shfl_xor(x, 8, 32));
        best[v] = x;
    }

    if (lm == 0) {
        int* dst = best_idx + (size_t)b * NP + (size_t)tile * 16 + 8 * half;
#pragma unroll
        for (int v = 0; v < 8; ++v) atomicMax(&dst[v], best[v]);
    }
}

// ---------------------------------------------------------------------------
// Kernel 2: gather + duplicated-K write. One 256-thread block per (b, j).
// Thread t: k4 = t%16 selects a float4 slot of the 64 identical k values,
// g = t/16 walks channels -> 16 lanes emit 256B contiguous stores per channel.
// ---------------------------------------------------------------------------
__global__ __launch_bounds__(256)
void gather_write(const float* __restrict__ pts,
                  const float* __restrict__ cents,
                  const float* __restrict__ feats,
                  const int* __restrict__ idx,
                  float* __restrict__ out) {
    const int bj = blockIdx.x;         // 0 .. BATCH*NP-1
    const int b  = bj >> 11;           // / NP
    const int t  = threadIdx.x;
    const int k4 = t & 15;
    const int g  = t >> 4;

    int iv = idx[bj];
    iv = iv < 0 ? 0 : iv;              // "no neighbor" -> index 0, like reference

    const float* p = pts + ((size_t)b * N_POINTS + iv) * 3;
    const float* c = cents + (size_t)bj * 3;

    for (int ch = g; ch < N_CH; ch += 16) {
        float val;
        if (ch < 3) {
            val = (p[ch] - c[ch]) * INV_RADIUS;
        } else {
            val = feats[((size_t)b * N_FEAT + (ch - 3)) * N_POINTS + iv];
        }
        float4 v4 = make_float4(val, val, val, val);
        float4* o = (float4*)out;
        o[(((size_t)b * N_CH + ch) * NP + (bj & (NP - 1))) * (K_CLUSTER / 4) + k4] = v4;
    }
}

extern "C" void kernel_launch(void* const* d_in, const int* in_sizes, int n_in,
                              void* d_out, int out_size, void* d_ws, size_t ws_size,
                              hipStream_t stream) {
    const float* pts   = (const float*)d_in[0];   // (8,16384,3) f32
    const float* cents = (const float*)d_in[1];   // (8,2048,3)  f32
    const float* feats = (const float*)d_in[2];   // (8,256,16384) f32
    float* out = (float*)d_out;                   // (8,259,2048,64) f32
    int* idxws = (int*)d_ws;                      // BATCH*NP ints

    // Init running-max index array to -1 (0xFF fill). Memset node is
    // graph-capture safe; re-done every launch -> deterministic.
    hipMemsetAsync(idxws, 0xFF, (size_t)BATCH * NP * sizeof(int), stream);

    dim3 g1((NP / 16) * CHUNKS, BATCH);           // 1024 x 8 = 8192 waves
    ball_query_wmma<<<g1, 32, 0, stream>>>(pts, cents, idxws);

    gather_write<<<BATCH * NP, 256, 0, stream>>>(pts, cents, feats, idxws, out);
}